// E2IMG_3092376453879
// MI455X (gfx1250) — compile-verified
//
#include <hip/hip_runtime.h>

// Image geometry is fixed by the reference.
#define IMG_H 720
#define IMG_W 1280

typedef float v4f __attribute__((ext_vector_type(4)));
typedef float v3f __attribute__((ext_vector_type(3)));

// ---------------------------------------------------------------------------
// Kernel 1: fill output with the "no event" pattern (255, 255, 510) per pixel.
// out layout: (B, H, W, 3) f32 -> flat channel = f % 3, value 510 iff ch==2.
// The pattern has period 3 floats == 3 float4s (12 floats), so each thread
// owns one full period starting at a multiple of 12: the three 128-bit store
// payloads are compile-time constants (no per-thread modulo at all).
// Regular (RT) stores on purpose: the 177MB output ~fits the 192MB L2, so
// these lines stay resident and the scatter pass hits L2 instead of HBM.
// ---------------------------------------------------------------------------
__global__ __launch_bounds__(256) void e2img_fill(float* __restrict__ out,
                                                  int n_f) {
  const v4f A = {255.0f, 255.0f, 510.0f, 255.0f};  // channels 0,1,2,0
  const v4f B = {255.0f, 510.0f, 255.0f, 255.0f};  // channels 1,2,0,1
  const v4f C = {510.0f, 255.0f, 255.0f, 510.0f};  // channels 2,0,1,2
  int base = (blockIdx.x * blockDim.x + threadIdx.x) * 12;
  if (base + 12 <= n_f) {
    *(v4f*)(out + base)     = A;
    *(v4f*)(out + base + 4) = B;
    *(v4f*)(out + base + 8) = C;
  } else if (base < n_f) {
    // Scalar tail (< 12 floats at the very end).
    for (int f = base; f < n_f; ++f)
      out[f] = ((f % 3) == 2) ? 510.0f : 255.0f;
  }
}

// ---------------------------------------------------------------------------
// Kernel 2: scatter events into the image.
// Event (t, xs, ys, p) of batch b = blockIdx.y (no per-thread divide):
//   ch0 = 255 - 255*p, ch1 = 255*p, ch2 = ch0 + ch1 = 255.
// - 128-bit non-temporal event loads (TH_NT): the 128MB stream is touch-once
//   and must not evict the L2-resident output.
// - global_prefetch_b8 runs 16K events (256KB) ahead of the block wavefront.
// - The 3-channel pixel write is a single GLOBAL_STORE_B96 via inline asm:
//   the address (12*pix bytes) is always 4B-aligned, which DWORD alignment
//   mode accepts for multi-DWORD global stores. One L2 request per event
//   instead of three.
// ---------------------------------------------------------------------------
__global__ __launch_bounds__(256) void e2img_scatter(const v4f* __restrict__ ev,
                                                     float* __restrict__ out,
                                                     int N) {
  int n = blockIdx.x * blockDim.x + threadIdx.x;
  if (n >= N) return;
  int b = blockIdx.y;
  size_t e = (size_t)b * (size_t)N + (size_t)n;

  // Prefetch the event stream well ahead (stays within this batch's slice).
  const int PF = 64 * 256;
  if (n + PF < N) {
    __builtin_prefetch((const void*)(ev + e + PF), 0, 0);
  }

  v4f v = __builtin_nontemporal_load(ev + e);

  int xi = (int)floorf(v.y);
  int yi = (int)floorf(v.z);
  float ch1 = 255.0f * v.w;   // 0 if p==0, 255 if p==1
  float ch0 = 255.0f - ch1;   // 255 if p==0, 0 if p==1

  size_t pix = ((size_t)b * IMG_H + (size_t)yi) * IMG_W + (size_t)xi;
  float* dst = out + pix * 3;

  v3f val;
  val.x = ch0;
  val.y = ch1;
  val.z = 255.0f;
  // Single 96-bit store: v[addr:addr+1] holds the 64-bit address, data in a
  // 3-VGPR tuple. (A plain vec3 store is unsafe: clang widens it to vec4.)
  asm volatile("global_store_b96 %0, %1, off"
               :
               : "v"(dst), "v"(val)
               : "memory");
}

extern "C" void kernel_launch(void* const* d_in, const int* in_sizes, int n_in,
                              void* d_out, int out_size, void* d_ws, size_t ws_size,
                              hipStream_t stream) {
  const float* x = (const float*)d_in[0];
  float* out = (float*)d_out;

  const int n_events = in_sizes[0] / 4;            // B * N
  const int per_img  = IMG_H * IMG_W * 3;
  int B = out_size / per_img;
  if (B < 1) B = 1;
  const int N = n_events / B;                       // events per batch

  {
    // One thread per 12-float period.
    int periods = (out_size + 11) / 12;
    int blocks  = (periods + 255) / 256;
    e2img_fill<<<blocks, 256, 0, stream>>>(out, out_size);
  }
  {
    dim3 grid((N + 255) / 256, B);
    e2img_scatter<<<grid, dim3(256), 0, stream>>>((const v4f*)x, out, N);
  }
}